// ListFold_84112639525724
// MI455X (gfx1250) — compile-verified
//
#include <hip/hip_runtime.h>
#include <hip/hip_bf16.h>

typedef __attribute__((ext_vector_type(2))) float v2f;
typedef __attribute__((ext_vector_type(8))) float v8f;

#define NCOL 512
#define PDIM 256
#define BLOCK 256

// One workgroup per batch row. 8 wave32 per workgroup; wave w owns u-blocks
// [32w,32w+16) and [32w+16,32w+32). row_sum computed as masked-exp-matrix @ ones
// via V_WMMA_F32_16X16X4_F32 accumulation. Loop bounds go through
// readfirstlane so loop control is scalar and EXEC provably stays all-1s
// around the WMMA (ISA requirement).
__global__ __launch_bounds__(BLOCK) void listfold_row_kernel(
    const float* __restrict__ scores, const int* __restrict__ rankings,
    float* __restrict__ row_loss)
{
    __shared__ float s_lds[NCOL];   // gathered scores for this row
    __shared__ float rs[PDIM];      // row sums, then suffix sums
    __shared__ float red[BLOCK];    // reduction buffer

    const int b    = blockIdx.x;
    const int tid  = threadIdx.x;
    const int lane = tid & 31;
    const int wave = tid >> 5;

    // Gather s[j] = scores[b, rankings[b, j]]
    for (int j = tid; j < NCOL; j += BLOCK) {
        const int idx = rankings[b * NCOL + j];
        s_lds[j] = scores[b * NCOL + idx];
    }
    __syncthreads();

    // All-ones B fragment (4x16), layout irrelevant since every element is 1.0
    v2f bones;
    bones.x = 1.0f;
    bones.y = 1.0f;

    // A-matrix 16x4 f32 layout (wave32): lanes 0-15 -> M=lane, K = {0,1} in
    // VGPR {0,1}; lanes 16-31 -> M=lane-16, K = {2,3}.
    const int m     = lane & 15;
    const int kbase = (lane >> 4) * 2;

    #pragma unroll
    for (int blk = 0; blk < 2; ++blk) {
        const int   u0  = wave * 32 + blk * 16;
        const int   u   = u0 + m;
        const float su  = s_lds[u];
        const int   umax = (NCOL - 1) - u;   // per-lane mask upper bound

        // Wave-uniform scalar loop bounds: v in [u0, 511-u0] grouped by 4.
        const int v0s  = __builtin_amdgcn_readfirstlane(u0);
        const int vhis = __builtin_amdgcn_readfirstlane((NCOL - 1) - u0);

        v8f acc = {};
        for (int v0 = v0s; v0 <= vhis; v0 += 4) {   // scalar loop control
            // Two consecutive K-elements per lane: 8B-aligned ds_load_b64
            const float2 sv = *reinterpret_cast<const float2*>(&s_lds[v0 + kbase]);
            v2f a;
            {
                const int v = v0 + kbase;
                float d = su - sv.x;
                d = fminf(fmaxf(d, -10.0f), 10.0f);
                const float e  = __expf(d);
                a[0] = ((v > u) && (v <= umax)) ? e : 0.0f;  // v_cndmask
            }
            {
                const int v = v0 + kbase + 1;
                float d = su - sv.y;
                d = fminf(fmaxf(d, -10.0f), 10.0f);
                const float e  = __expf(d);
                a[1] = ((v > u) && (v <= umax)) ? e : 0.0f;
            }
            // D = A x ones + C : every column of D = per-row masked sum
            acc = __builtin_amdgcn_wmma_f32_16x16x4_f32(
                false, a, false, bones, (short)0, acc, false, false);
        }

        // C/D 16x16 f32 layout: lane 0 VGPR r -> M=r (N=0); lane 16 -> M=8+r
        if (lane == 0) {
            #pragma unroll
            for (int r = 0; r < 8; ++r) rs[u0 + r] = acc[r];
        }
        if (lane == 16) {
            #pragma unroll
            for (int r = 0; r < 8; ++r) rs[u0 + 8 + r] = acc[r];
        }
    }
    __syncthreads();

    // Suffix sum: den[u] = EPS + sum_{k>=u} row_sum[k]  (256 serial adds, tiny)
    if (tid == 0) {
        float run = 0.0f;
        for (int i = PDIM - 1; i >= 0; --i) {
            run += rs[i];
            rs[i] = run;
        }
    }
    __syncthreads();

    // Per-u term: log(den) - clip(s_u - s_{n-1-u});  loss_b = sum of terms
    {
        const int u = tid;
        const float den = 1e-10f + rs[u];
        float c = s_lds[u] - s_lds[NCOL - 1 - u];
        c = fminf(fmaxf(c, -10.0f), 10.0f);
        red[u] = __logf(den) - c;
    }
    __syncthreads();
    for (int off = BLOCK / 2; off > 0; off >>= 1) {
        if (tid < off) red[tid] += red[tid + off];
        __syncthreads();
    }
    if (tid == 0) row_loss[b] = red[0];
}

// Reduce B per-row losses -> mean, single block.
__global__ __launch_bounds__(BLOCK) void listfold_final_kernel(
    const float* __restrict__ row_loss, float* __restrict__ out, int B)
{
    __shared__ float red[BLOCK];
    const int tid = threadIdx.x;
    float acc = 0.0f;
    for (int i = tid; i < B; i += BLOCK) acc += row_loss[i];
    red[tid] = acc;
    __syncthreads();
    for (int off = BLOCK / 2; off > 0; off >>= 1) {
        if (tid < off) red[tid] += red[tid + off];
        __syncthreads();
    }
    if (tid == 0) out[0] = red[0] / (float)B;
}

extern "C" void kernel_launch(void* const* d_in, const int* in_sizes, int n_in,
                              void* d_out, int out_size, void* d_ws, size_t ws_size,
                              hipStream_t stream)
{
    const float* scores   = (const float*)d_in[0];
    const int*   rankings = (const int*)d_in[1];
    float*       out      = (float*)d_out;
    float*       row_loss = (float*)d_ws;   // B floats of scratch

    const int B = in_sizes[0] / NCOL;       // 256 for the reference shapes

    listfold_row_kernel<<<B, BLOCK, 0, stream>>>(scores, rankings, row_loss);
    listfold_final_kernel<<<1, BLOCK, 0, stream>>>(row_loss, out, B);
}